// IntNBitTableBatchedEmbeddingBagsCodegen_32289564131528
// MI455X (gfx1250) — compile-verified
//
#include <hip/hip_runtime.h>

// Problem constants (match reference)
#define T_TABLES 8
#define E_ROWS   500000
#define D_DIM    128
#define B_BATCH  2048
#define L_BAG    50
#define ROW_BYTES 80

#define WPB 8                                   // waves per block (wave32)
#define LDS_ROW_STRIDE 88                       // pad 80 -> 88 B: half-wave read windows hit disjoint banks
#define ROW_REGION (L_BAG * LDS_ROW_STRIDE)     // 4400 B
#define WAVE_LDS   4608                         // 4400 + 200 sw table, padded

typedef _Float16 h8 __attribute__((ext_vector_type(8)));
typedef _Float16 h2 __attribute__((ext_vector_type(2)));

// 2^24: cancels the f16-denormal scaling of raw nibble bits (n as f16 bits == n * 2^-24)
#define NIB_SCALE 16777216.0f

__global__ __launch_bounds__(WPB * 32)
void embedding_bag_4bit_kernel(const int* __restrict__ indices,
                               const float* __restrict__ psw,
                               const unsigned char* __restrict__ wq,
                               _Float16* __restrict__ out)
{
    __shared__ __attribute__((aligned(16))) unsigned char smem[WPB * WAVE_LDS];

    const int lane = threadIdx.x & 31;
    const int wave = threadIdx.x >> 5;
    const int bag  = blockIdx.x * WPB + wave;      // bag = t*B + b (matches (T,B,L) flatten)
    const int t    = bag >> 11;                    // / B_BATCH
    const int b    = bag & (B_BATCH - 1);

    unsigned char* wbase = smem + wave * WAVE_LDS;
    const unsigned lds_base = (unsigned)(uintptr_t)wbase;   // low 32b of flat LDS addr = LDS offset

    const int sbase = bag * L_BAG;
    const unsigned char* table = wq + (size_t)t * E_ROWS * ROW_BYTES;

    // ---- Stage 1: async DMA gather of 50 rows x 80B into LDS ------------
    // lane <-> row; IOFFSET advances BOTH the LDS dst and the global src.
    #pragma unroll
    for (int k = 0; k < 2; ++k) {
        int r = lane + k * 32;
        if (r < L_BAG) {
            int row = indices[sbase + r];
            const unsigned char* src = table + (size_t)row * ROW_BYTES;
            unsigned dst = lds_base + r * LDS_ROW_STRIDE;
            asm volatile("global_load_async_to_lds_b128 %0, %1, off\n\t"
                         "global_load_async_to_lds_b128 %0, %1, off offset:16\n\t"
                         "global_load_async_to_lds_b128 %0, %1, off offset:32\n\t"
                         "global_load_async_to_lds_b128 %0, %1, off offset:48\n\t"
                         "global_load_async_to_lds_b128 %0, %1, off offset:64"
                         :: "v"(dst), "v"(src) : "memory");
        }
    }

    // Overlap: fetch per-sample weights while the DMA is in flight.
    float w0 = psw[sbase + lane];                                    // rows 0..31
    float w1 = (lane < L_BAG - 32) ? psw[sbase + 32 + lane] : 0.f;   // rows 32..49

    asm volatile("s_wait_asynccnt 0x0" ::: "memory");

    // ---- Stage 2: swtab[r] = scale*w*2^24 ; bias term wave-reduced ------
    float* swtab = (float*)(wbase + ROW_REGION);
    float bw = 0.f;
    {
        unsigned hdr = *(const unsigned*)(wbase + lane * LDS_ROW_STRIDE);
        union { unsigned short u; _Float16 f; } cs, cb;
        cs.u = (unsigned short)(hdr & 0xFFFFu);
        cb.u = (unsigned short)(hdr >> 16);
        swtab[lane] = (float)cs.f * w0 * NIB_SCALE;
        bw += (float)cb.f * w0;
    }
    if (lane < L_BAG - 32) {
        unsigned hdr = *(const unsigned*)(wbase + (lane + 32) * LDS_ROW_STRIDE);
        union { unsigned short u; _Float16 f; } cs, cb;
        cs.u = (unsigned short)(hdr & 0xFFFFu);
        cb.u = (unsigned short)(hdr >> 16);
        swtab[lane + 32] = (float)cs.f * w1 * NIB_SCALE;
        bw += (float)cb.f * w1;
    }
    #pragma unroll
    for (int m = 16; m >= 1; m >>= 1) bw += __shfl_xor(bw, m, 32);
    asm volatile("s_wait_dscnt 0x0" ::: "memory");  // swtab visible (same-wave DS in order)

    // ---- Stage 3: dequant + weighted pooling ----------------------------
    // lane (lane&15) owns dims 8*l..8*l+7; half-wave 0: rows 0..24, half 1: 25..49.
    // Nibbles become raw f16 denormals; v_fma_mix_f32 promotes exactly and
    // accumulates in full f32: acc += (n*2^-24) * (sw*2^24) == n*sw exactly.
    const int half = lane >> 4;
    const int lsub = lane & 15;
    const unsigned char* mydata = wbase + 4 + lsub * 4;
    const int rbase = half * 25;

    float a0=0.f,a1=0.f,a2=0.f,a3=0.f,a4=0.f,a5=0.f,a6=0.f,a7=0.f;
    #pragma unroll
    for (int i = 0; i < 25; ++i) {
        const int r = rbase + i;
        float sw24 = swtab[r];                                        // b32 broadcast (vectorized)
        unsigned h = *(const unsigned*)(mydata + r * LDS_ROW_STRIDE); // conflict-free b32
        h2 q0 = __builtin_bit_cast(h2, (unsigned)( h        & 0x000F000Fu)); // (n0, n4)
        h2 q1 = __builtin_bit_cast(h2, (unsigned)((h >> 4)  & 0x000F000Fu)); // (n1, n5)
        h2 q2 = __builtin_bit_cast(h2, (unsigned)((h >> 8)  & 0x000F000Fu)); // (n2, n6)
        h2 q3 = __builtin_bit_cast(h2, (unsigned)((h >> 12) & 0x000F000Fu)); // (n3, n7)
        a0 = fmaf((float)q0.x, sw24, a0);   // v_fma_mix_f32, lo half
        a1 = fmaf((float)q1.x, sw24, a1);
        a2 = fmaf((float)q2.x, sw24, a2);
        a3 = fmaf((float)q3.x, sw24, a3);
        a4 = fmaf((float)q0.y, sw24, a4);   // v_fma_mix_f32, hi half
        a5 = fmaf((float)q1.y, sw24, a5);
        a6 = fmaf((float)q2.y, sw24, a6);
        a7 = fmaf((float)q3.y, sw24, a7);
    }

    // cross-half reduction, then pooled bias term
    a0 += __shfl_xor(a0, 16, 32); a1 += __shfl_xor(a1, 16, 32);
    a2 += __shfl_xor(a2, 16, 32); a3 += __shfl_xor(a3, 16, 32);
    a4 += __shfl_xor(a4, 16, 32); a5 += __shfl_xor(a5, 16, 32);
    a6 += __shfl_xor(a6, 16, 32); a7 += __shfl_xor(a7, 16, 32);

    if (lane < 16) {
        h8 o;
        o.s0 = (_Float16)(a0 + bw); o.s1 = (_Float16)(a1 + bw);
        o.s2 = (_Float16)(a2 + bw); o.s3 = (_Float16)(a3 + bw);
        o.s4 = (_Float16)(a4 + bw); o.s5 = (_Float16)(a5 + bw);
        o.s6 = (_Float16)(a6 + bw); o.s7 = (_Float16)(a7 + bw);
        // out[b][t*128 + 8*lsub .. +7], fp16: 16B/lane, 256B contiguous per wave
        *(h8*)(out + (size_t)b * (T_TABLES * D_DIM) + t * D_DIM + lsub * 8) = o;
    }
}

extern "C" void kernel_launch(void* const* d_in, const int* in_sizes, int n_in,
                              void* d_out, int out_size, void* d_ws, size_t ws_size,
                              hipStream_t stream) {
    (void)n_in; (void)out_size; (void)d_ws; (void)ws_size;
    const int*           indices = (const int*)d_in[0];
    // d_in[1] = offsets (unused: bags are uniform length L)
    const float*         pswp    = (const float*)d_in[2];
    const unsigned char* wqp     = (const unsigned char*)d_in[3];
    _Float16*            outp    = (_Float16*)d_out;

    const int bags   = in_sizes[0] / L_BAG;        // T*B = 16384
    const int blocks = bags / WPB;                 // 2048
    embedding_bag_4bit_kernel<<<blocks, WPB * 32, 0, stream>>>(indices, pswp, wqp, outp);
}